// DMETM_10746008175092
// MI455X (gfx1250) — compile-verified
//
#include <hip/hip_runtime.h>
#include <hip/hip_bf16.h>
#include <math.h>

// Problem constants (from reference)
#define KK 50     // topics
#define TT 50     // timesteps
#define VV 3000   // vocab
#define LL 300    // latent
#define THH 800   // theta hidden
#define EHH 200   // eta hidden
#define SS 8      // sources
#define DD 256    // docs
#define CHUNK 32  // docs per nll chunk

typedef __attribute__((ext_vector_type(2))) float v2f;
typedef __attribute__((ext_vector_type(8))) float v8f;

// ---------------------------------------------------------------------------
// Interior fp32 WMMA GEMM, NO bounds checks. One wave computes a 16x64 strip
// (4 x 16x16 tiles sharing one A fragment): 4 WMMAs per K-step of 4.
// Requires: 16 | m-extent, 64 | n-extent covered by the grid; Kd >= 4.
//   TRANS_B=0 : B is KxN row-major;  TRANS_B=1 : B is NxK row-major.
// ---------------------------------------------------------------------------
template<bool TRANS_B, int ACT>
__global__ __launch_bounds__(32)
void wmma_gemm_quad(const float* __restrict__ A, int lda,
                    const float* __restrict__ B, int ldb,
                    const float* __restrict__ bias,
                    float* __restrict__ C, int ldc, int Kd) {
    const int lane  = threadIdx.x & 31;
    const int half  = lane >> 4;
    const int idx16 = lane & 15;
    const int m0 = blockIdx.y * 16;
    const int n0 = blockIdx.x * 64;

    // Per-lane fragment base pointers (A: row major, K pair chosen by half)
    const float* Ap = A + (long)(m0 + idx16) * lda + 2 * half;
    const float* Bp[4];
#pragma unroll
    for (int j = 0; j < 4; ++j) {
        const int n = n0 + 16 * j + idx16;
        Bp[j] = TRANS_B ? (B + (long)n * ldb + 2 * half)
                        : (B + (long)(2 * half) * ldb + n);
    }

    v8f acc[4] = {};
    const int kmain = Kd & ~3;
#pragma unroll 2
    for (int k = 0; k < kmain; k += 4) {
        const v2f a = *(const v2f*)(Ap + k);   // 8B aligned (all lda even)
#pragma unroll
        for (int j = 0; j < 4; ++j) {
            v2f b;
            if (TRANS_B) {
                b = *(const v2f*)(Bp[j] + k);
            } else {
                b.x = Bp[j][(long)k * ldb];
                b.y = Bp[j][(long)(k + 1) * ldb];
            }
            acc[j] = __builtin_amdgcn_wmma_f32_16x16x4_f32(
                         false, a, false, b, (short)0, acc[j], false, false);
        }
    }
    // Scalar K remainder in accumulator layout: acc[j][i] = C[m0+i+8*half][n]
    for (int k = kmain; k < Kd; ++k) {
#pragma unroll
        for (int j = 0; j < 4; ++j) {
            const int n = n0 + 16 * j + idx16;
            const float bv = TRANS_B ? B[(long)n * ldb + k] : B[(long)k * ldb + n];
#pragma unroll
            for (int i = 0; i < 8; ++i)
                acc[j][i] += A[(long)(m0 + i + 8 * half) * lda + k] * bv;
        }
    }
#pragma unroll
    for (int j = 0; j < 4; ++j) {
        const int n = n0 + 16 * j + idx16;
        const float bb = (bias != nullptr) ? bias[n] : 0.f;
#pragma unroll
        for (int i = 0; i < 8; ++i) {
            float v = acc[j][i] + bb;
            if (ACT == 1) v = fmaxf(v, 0.f);
            C[(long)(m0 + i + 8 * half) * ldc + n] = v;
        }
    }
}

// ---------------------------------------------------------------------------
// Guarded single-tile WMMA GEMM for edge strips (m_base/n_base offsets).
// ---------------------------------------------------------------------------
template<bool TRANS_B, int ACT>
__global__ __launch_bounds__(32)
void wmma_gemm_guard(const float* __restrict__ A, int lda,
                     const float* __restrict__ B, int ldb,
                     const float* __restrict__ bias,
                     float* __restrict__ C, int ldc,
                     int M, int N, int Kd, int m_base, int n_base) {
    const int lane  = threadIdx.x & 31;
    const int half  = lane >> 4;
    const int idx16 = lane & 15;
    const int m0 = m_base + blockIdx.y * 16;
    const int n0 = n_base + blockIdx.x * 16;
    const int am = m0 + idx16;
    const int bn = n0 + idx16;

    v8f acc = {};
    const int kmain = Kd & ~3;
    for (int k = 0; k < kmain; k += 4) {
        const int ka = k + 2 * half;
        v2f a, b;
        a.x = (am < M) ? A[(long)am * lda + ka]     : 0.f;
        a.y = (am < M) ? A[(long)am * lda + ka + 1] : 0.f;
        if (TRANS_B) {
            b.x = (bn < N) ? B[(long)bn * ldb + ka]     : 0.f;
            b.y = (bn < N) ? B[(long)bn * ldb + ka + 1] : 0.f;
        } else {
            b.x = (bn < N) ? B[(long)ka * ldb + bn]       : 0.f;
            b.y = (bn < N) ? B[(long)(ka + 1) * ldb + bn] : 0.f;
        }
        acc = __builtin_amdgcn_wmma_f32_16x16x4_f32(
                  false, a, false, b, (short)0, acc, false, false);
    }
    for (int k = kmain; k < Kd; ++k) {
        float bv = 0.f;
        if (bn < N) bv = TRANS_B ? B[(long)bn * ldb + k] : B[(long)k * ldb + bn];
#pragma unroll
        for (int i = 0; i < 8; ++i) {
            const int m = m0 + i + 8 * half;
            const float av = (m < M) ? A[(long)m * lda + k] : 0.f;
            acc[i] += av * bv;
        }
    }
    const float bb = (bias != nullptr && bn < N) ? bias[bn] : 0.f;
#pragma unroll
    for (int i = 0; i < 8; ++i) {
        const int m = m0 + i + 8 * half;
        if (m < M && bn < N) {
            float v = acc[i] + bb;
            if (ACT == 1) v = fmaxf(v, 0.f);
            C[(long)m * ldc + bn] = v;
        }
    }
}

// Host-side dispatcher: interior quad tiles + guarded edge strips.
template<bool TB, int ACT>
static void launch_gemm(const float* A, int lda, const float* B, int ldb,
                        const float* bias, float* C, int ldc,
                        int M, int N, int Kd, hipStream_t stream) {
    const int Mf = M & ~15;   // rows covered by full tiles
    const int N4 = N & ~63;   // cols covered by quad strips
    if (Mf > 0 && N4 > 0)
        wmma_gemm_quad<TB, ACT><<<dim3(N4 / 64, Mf / 16), 32, 0, stream>>>(
            A, lda, B, ldb, bias, C, ldc, Kd);
    if (N4 < N)   // right strip: all rows, cols [N4, N)
        wmma_gemm_guard<TB, ACT><<<dim3((N - N4 + 15) / 16, (M + 15) / 16), 32, 0, stream>>>(
            A, lda, B, ldb, bias, C, ldc, M, N, Kd, 0, N4);
    if (Mf < M && N4 > 0)   // bottom strip: rows [Mf, M), cols [0, N4)
        wmma_gemm_guard<TB, ACT><<<dim3(N4 / 16, 1), 32, 0, stream>>>(
            A, lda, B, ldb, bias, C, ldc, M, N, Kd, Mf, 0);
}

// ---------------------------------------------------------------------------
// KL(alpha): sum over K,T,L of Gaussian KL terms -> d_out[1]
// ---------------------------------------------------------------------------
__global__ void kl_alpha_kernel(const float* __restrict__ mu,
                                const float* __restrict__ ls,
                                float* __restrict__ d_out, float log_delta) {
    const int total = KK * TT * LL;
    int idx = blockIdx.x * blockDim.x + threadIdx.x;
    float acc = 0.f;
    for (int i = idx; i < total; i += gridDim.x * blockDim.x) {
        int l  = i % LL;
        int kt = i / LL;
        int t  = kt % TT;
        int k  = kt / TT;
        float qm = mu[i], ql = ls[i];
        if (t == 0) {
            acc += 0.5f * (__expf(ql) + qm * qm - 1.0f - ql);
        } else {
            float pm = mu[(k * TT + (t - 1)) * LL + l];
            float dif = qm - pm;
            acc += 0.5f * ((__expf(ql) + dif * dif) / (__expf(log_delta) + 1e-6f)
                           - 1.0f + log_delta - ql);
        }
    }
    __shared__ float red[256];
    red[threadIdx.x] = acc;
    __syncthreads();
    for (int s = 128; s > 0; s >>= 1) {
        if (threadIdx.x < s) red[threadIdx.x] += red[threadIdx.x + s];
        __syncthreads();
    }
    if (threadIdx.x == 0) atomicAdd(&d_out[1], red[0]);
}

// ---------------------------------------------------------------------------
// Two stacked LSTM layers, sequential over T. One block, 800 threads (=4*EH).
// ---------------------------------------------------------------------------
__global__ void lstm2_kernel(const float* __restrict__ x_em,
                             const float* __restrict__ Wih0, const float* __restrict__ Whh0,
                             const float* __restrict__ bl0,
                             const float* __restrict__ Wih1, const float* __restrict__ Whh1,
                             const float* __restrict__ bl1,
                             float* __restrict__ hs0, float* __restrict__ out) {
    const int j = threadIdx.x;                 // 0 .. 4*EH-1
    __shared__ float h[EHH], c[EHH], g[4 * EHH], xt[EHH];

    for (int layer = 0; layer < 2; ++layer) {
        const float* Wih = layer ? Wih1 : Wih0;
        const float* Whh = layer ? Whh1 : Whh0;
        const float* bl  = layer ? bl1  : bl0;
        const float* xin = layer ? hs0  : x_em;
        float* hout      = layer ? out  : hs0;
        if (j < EHH) { h[j] = 0.f; c[j] = 0.f; }
        __syncthreads();
        for (int t = 0; t < TT; ++t) {
            if (j < EHH) xt[j] = xin[t * EHH + j];
            __syncthreads();
            float gv = bl[j];
            for (int e = 0; e < EHH; ++e)
                gv += xt[e] * Wih[e * 4 * EHH + j] + h[e] * Whh[e * 4 * EHH + j];
            g[j] = gv;
            __syncthreads();
            if (j < EHH) {
                float ig = 1.f / (1.f + __expf(-g[j]));
                float fg = 1.f / (1.f + __expf(-g[EHH + j]));
                float gg = tanhf(g[2 * EHH + j]);
                float og = 1.f / (1.f + __expf(-g[3 * EHH + j]));
                float c2 = fg * c[j] + ig * gg;
                float h2 = og * tanhf(c2);
                c[j] = c2; h[j] = h2;
                hout[t * EHH + j] = h2;
            }
            __syncthreads();
        }
        __syncthreads();
    }
}

// ---------------------------------------------------------------------------
// eta scan: sequential over T, single block of 64 threads. Writes etas and
// kl_eta -> d_out[2].
// ---------------------------------------------------------------------------
__global__ void estep_kernel(const float* __restrict__ out_l1,
                             const float* __restrict__ W_mu_e, const float* __restrict__ b_mu_e,
                             const float* __restrict__ W_ls_e, const float* __restrict__ b_ls_e,
                             float* __restrict__ etas, float* __restrict__ d_out,
                             float log_delta) {
    const int j = threadIdx.x;
    __shared__ float cat[EHH + KK];
    __shared__ float eta_prev[KK];
    __shared__ float klacc[KK];
    if (j < KK) eta_prev[j] = 0.f;
    __syncthreads();
    float kl_total = 0.f;
    for (int t = 0; t < TT; ++t) {
        for (int i = j; i < EHH; i += blockDim.x) cat[i] = out_l1[t * EHH + i];
        if (j < KK) cat[EHH + j] = eta_prev[j];
        __syncthreads();
        const float pls_t = (t == 0) ? 0.f : log_delta;
        if (j < KK) {
            float mu = b_mu_e[j], ls = b_ls_e[j];
            for (int i = 0; i < EHH + KK; ++i) {
                mu += cat[i] * W_mu_e[i * KK + j];
                ls += cat[i] * W_ls_e[i * KK + j];
            }
            float ep  = eta_prev[j];
            float dif = mu - ep;
            klacc[j] = 0.5f * ((__expf(ls) + dif * dif) / (__expf(pls_t) + 1e-6f)
                               - 1.0f + pls_t - ls);
            etas[t * KK + j] = mu;
        }
        __syncthreads();
        if (j < KK) eta_prev[j] = etas[t * KK + j];
        if (j == 0) for (int q = 0; q < KK; ++q) kl_total += klacc[q];
        __syncthreads();
    }
    if (j == 0) d_out[2] = kl_total;
}

// ---------------------------------------------------------------------------
// Prep: eta_td gather, Acat = [bows | eta_td], Abig[dK, l] = alpha*lam gather.
// ---------------------------------------------------------------------------
__global__ void prep_kernel(const float* __restrict__ bows,
                            const float* __restrict__ etas,
                            const int* __restrict__ times,
                            const int* __restrict__ sources,
                            const float* __restrict__ alpha,
                            const float* __restrict__ lam,
                            float* __restrict__ eta_td,
                            float* __restrict__ Acat,
                            float* __restrict__ Abig) {
    const long stride = (long)gridDim.x * blockDim.x;
    const long idx = (long)blockIdx.x * blockDim.x + threadIdx.x;
    for (long i = idx; i < (long)DD * KK; i += stride) {
        int d = (int)(i / KK), k = (int)(i % KK);
        eta_td[i] = etas[times[d] * KK + k];
    }
    const int W = VV + KK;
    for (long i = idx; i < (long)DD * W; i += stride) {
        int d = (int)(i / W), col = (int)(i % W);
        Acat[i] = (col < VV) ? bows[(long)d * VV + col]
                             : etas[times[d] * KK + (col - VV)];
    }
    for (long i = idx; i < (long)DD * KK * LL; i += stride) {
        int l = (int)(i % LL);
        long dk = i / LL;
        int k = (int)(dk % KK), d = (int)(dk / KK);
        Abig[i] = alpha[((long)k * TT + times[d]) * LL + l]
                * lam[(long)sources[d] * LL + l];
    }
}

// ---------------------------------------------------------------------------
// theta = softmax(mu_th) over K; kl_theta -> d_out[3]. One block per doc.
// ---------------------------------------------------------------------------
__global__ void theta_kernel(const float* __restrict__ mu_th,
                             const float* __restrict__ ls_th,
                             const float* __restrict__ eta_td,
                             float* __restrict__ theta,
                             float* __restrict__ d_out) {
    const int d = blockIdx.x, j = threadIdx.x;
    __shared__ float red[64];
    if (j == 0) {
        float mx = mu_th[d * KK];
        for (int k = 1; k < KK; ++k) mx = fmaxf(mx, mu_th[d * KK + k]);
        float s = 0.f;
        for (int k = 0; k < KK; ++k) {
            float e = __expf(mu_th[d * KK + k] - mx);
            theta[d * KK + k] = e; s += e;
        }
        float inv = 1.f / s;
        for (int k = 0; k < KK; ++k) theta[d * KK + k] *= inv;
    }
    float term = 0.f;
    if (j < KK) {
        float mu = mu_th[d * KK + j], ls = ls_th[d * KK + j];
        float dif = mu - eta_td[d * KK + j];
        term = 0.5f * ((__expf(ls) + dif * dif) / (1.0f + 1e-6f) - 1.0f - ls);
    }
    red[j] = term;
    __syncthreads();
    for (int s = 32; s > 0; s >>= 1) {
        if (j < s) red[j] += red[j + s];
        __syncthreads();
    }
    if (j == 0) atomicAdd(&d_out[3], red[0]);
}

// ---------------------------------------------------------------------------
// Softmax stats over V for each logit row.
// ---------------------------------------------------------------------------
__global__ void softstat_kernel(const float* __restrict__ logit,
                                float* __restrict__ mx, float* __restrict__ sm) {
    const int r = blockIdx.x, j = threadIdx.x;
    __shared__ float red[256];
    float m = -3.4e38f;
    for (int v = j; v < VV; v += 256) m = fmaxf(m, logit[(long)r * VV + v]);
    red[j] = m; __syncthreads();
    for (int s = 128; s > 0; s >>= 1) {
        if (j < s) red[j] = fmaxf(red[j], red[j + s]);
        __syncthreads();
    }
    const float M = red[0];
    __syncthreads();
    float s = 0.f;
    for (int v = j; v < VV; v += 256) s += __expf(logit[(long)r * VV + v] - M);
    red[j] = s; __syncthreads();
    for (int st = 128; st > 0; st >>= 1) {
        if (j < st) red[j] += red[j + st];
        __syncthreads();
    }
    if (j == 0) { mx[r] = M; sm[r] = red[0]; }
}

// ---------------------------------------------------------------------------
// NLL for one doc chunk: lik[d,v] = sum_k theta[d,k]*softmax(logit)[k,v];
// accumulate -log(lik+1e-6)*bows into d_out[0].
// ---------------------------------------------------------------------------
__global__ void nll_kernel(const float* __restrict__ logit,
                           const float* __restrict__ mx, const float* __restrict__ sm,
                           const float* __restrict__ theta, const float* __restrict__ bows,
                           int d0, float* __restrict__ d_out) {
    const int dc = blockIdx.x;          // doc within chunk
    const int d  = d0 + dc;
    const int j  = threadIdx.x;
    __shared__ float w[KK], mm[KK], red[256];
    if (j < KK) {
        int r = dc * KK + j;
        w[j]  = theta[d * KK + j] / sm[r];
        mm[j] = mx[r];
    }
    __syncthreads();
    float acc = 0.f;
    for (int v = j + blockIdx.y * 256; v < VV; v += 256 * gridDim.y) {
        float lik = 0.f;
        for (int k = 0; k < KK; ++k)
            lik += w[k] * __expf(logit[(long)(dc * KK + k) * VV + v] - mm[k]);
        acc += -__logf(lik + 1e-6f) * bows[(long)d * VV + v];
    }
    red[j] = acc; __syncthreads();
    for (int s = 128; s > 0; s >>= 1) {
        if (j < s) red[j] += red[j + s];
        __syncthreads();
    }
    if (j == 0) atomicAdd(&d_out[0], red[0]);
}

// ---------------------------------------------------------------------------
extern "C" void kernel_launch(void* const* d_in, const int* in_sizes, int n_in,
                              void* d_out, int out_size, void* d_ws, size_t ws_size,
                              hipStream_t stream) {
    (void)in_sizes; (void)n_in; (void)out_size; (void)ws_size;
    const float* bows     = (const float*)d_in[0];
    const float* rnn_inp  = (const float*)d_in[1];
    const int*   times    = (const int*)d_in[2];
    const int*   sources  = (const int*)d_in[3];
    const float* rho      = (const float*)d_in[4];
    const float* lam      = (const float*)d_in[5];
    const float* mu_q_a   = (const float*)d_in[6];
    const float* ls_q_a   = (const float*)d_in[7];
    const float* W_t1     = (const float*)d_in[8];
    const float* b_t1     = (const float*)d_in[9];
    const float* W_t2     = (const float*)d_in[10];
    const float* b_t2     = (const float*)d_in[11];
    const float* W_mu_th  = (const float*)d_in[12];
    const float* b_mu_th  = (const float*)d_in[13];
    const float* W_ls_th  = (const float*)d_in[14];
    const float* b_ls_th  = (const float*)d_in[15];
    const float* W_em     = (const float*)d_in[16];
    const float* b_em     = (const float*)d_in[17];
    const float* Wih0     = (const float*)d_in[18];
    const float* Whh0     = (const float*)d_in[19];
    const float* bl0      = (const float*)d_in[20];
    const float* Wih1     = (const float*)d_in[21];
    const float* Whh1     = (const float*)d_in[22];
    const float* bl1      = (const float*)d_in[23];
    const float* W_mu_e   = (const float*)d_in[24];
    const float* b_mu_e   = (const float*)d_in[25];
    const float* W_ls_e   = (const float*)d_in[26];
    const float* b_ls_e   = (const float*)d_in[27];
    float* out = (float*)d_out;

    const float log_delta = logf(0.005f);

    // Workspace carve-up (floats) — all offsets even (for 8B-aligned v2f loads)
    float* ws      = (float*)d_ws;
    float* x_em    = ws;                         // 50*200      = 10,000
    float* hs0     = x_em    + 10000;            // 10,000
    float* out_l1  = hs0     + 10000;            // 10,000
    float* etas    = out_l1  + 10000;            // 50*50       = 2,500
    float* eta_td  = etas    + 2500;             // 256*50      = 12,800
    float* Acat    = eta_td  + 12800;            // 256*3050    = 780,800
    float* h1      = Acat    + 780800;           // 256*800     = 204,800
    float* h2      = h1      + 204800;           // 204,800
    float* mu_th   = h2      + 204800;           // 12,800
    float* ls_th   = mu_th   + 12800;            // 12,800
    float* theta   = ls_th   + 12800;            // 12,800
    float* Abig    = theta   + 12800;            // 12800*300   = 3,840,000
    float* logit   = Abig    + 3840000;          // 1600*3000   = 4,800,000
    float* mxb     = logit   + 4800000;          // 1,600
    float* smb     = mxb     + 1600;             // 1,600

    hipMemsetAsync(d_out, 0, 4 * sizeof(float), stream);

    // KL(alpha) -> out[1]
    kl_alpha_kernel<<<512, 256, 0, stream>>>(mu_q_a, ls_q_a, out, log_delta);

    // x_em = rnn_inp @ W_em + b_em    (50x3000 @ 3000x200)
    launch_gemm<false, 0>(rnn_inp, VV, W_em, EHH, b_em, x_em, EHH,
                          TT, EHH, VV, stream);

    // Two LSTM layers (sequential)
    lstm2_kernel<<<1, 4 * EHH, 0, stream>>>(x_em, Wih0, Whh0, bl0,
                                            Wih1, Whh1, bl1, hs0, out_l1);

    // eta scan -> etas, out[2]
    estep_kernel<<<1, 64, 0, stream>>>(out_l1, W_mu_e, b_mu_e, W_ls_e, b_ls_e,
                                       etas, out, log_delta);

    // Gathers: eta_td, Acat, Abig
    prep_kernel<<<2048, 256, 0, stream>>>(bows, etas, times, sources,
                                          mu_q_a, lam, eta_td, Acat, Abig);

    // h1 = relu(Acat @ W_t1 + b_t1)   (256x3050 @ 3050x800)
    launch_gemm<false, 1>(Acat, VV + KK, W_t1, THH, b_t1, h1, THH,
                          DD, THH, VV + KK, stream);
    // h2 = relu(h1 @ W_t2 + b_t2)     (256x800 @ 800x800)
    launch_gemm<false, 1>(h1, THH, W_t2, THH, b_t2, h2, THH,
                          DD, THH, THH, stream);
    // mu_th / ls_th                   (256x800 @ 800x50)
    launch_gemm<false, 0>(h2, THH, W_mu_th, KK, b_mu_th, mu_th, KK,
                          DD, KK, THH, stream);
    launch_gemm<false, 0>(h2, THH, W_ls_th, KK, b_ls_th, ls_th, KK,
                          DD, KK, THH, stream);

    // theta softmax + KL(theta) -> out[3]
    theta_kernel<<<DD, 64, 0, stream>>>(mu_th, ls_th, eta_td, theta, out);

    // Big contraction + softmax + NLL, chunked over docs
    const int rows = CHUNK * KK; // 1600 logit rows per chunk
    for (int c = 0; c < DD / CHUNK; ++c) {
        const int d0 = c * CHUNK;
        // logit = Abig_chunk (1600x300) @ rho^T (rho is VxL row-major)
        launch_gemm<true, 0>(Abig + (long)d0 * KK * LL, LL, rho, LL, nullptr,
                             logit, VV, rows, VV, LL, stream);
        softstat_kernel<<<rows, 256, 0, stream>>>(logit, mxb, smb);
        dim3 g2(CHUNK, 4);
        nll_kernel<<<g2, 256, 0, stream>>>(logit, mxb, smb, theta, bows, d0, out);
    }
}